// Nested_Res2Net_TDNN_4784593568132
// MI455X (gfx1250) — compile-verified
//
#include <hip/hip_runtime.h>
#include <hip/hip_bf16.h>

typedef float v2f __attribute__((ext_vector_type(2)));
typedef float v8f __attribute__((ext_vector_type(8)));

#define T600 600
#define NBATCH 32
#define EPSF 1e-5f

// ---------------------------------------------------------------------------
// k_prep: x_in = x[:, 128i:128(i+1), :] (+ carry if i>0)
// ---------------------------------------------------------------------------
__global__ __launch_bounds__(256) void k_prep(const float* __restrict__ x,
                                              const float* __restrict__ carry,
                                              float* __restrict__ x_in,
                                              int iblk, int use_carry) {
  size_t idx = (size_t)blockIdx.x * 256 + threadIdx.x;
  const size_t NTOT = (size_t)NBATCH * 128 * T600;
  if (idx >= NTOT) return;
  size_t b = idx / (128 * T600);
  size_t rem = idx - b * (128 * T600);
  float v = x[b * 1024 * T600 + (size_t)iblk * 128 * T600 + rem];
  if (use_carry) v += carry[idx];
  x_in[idx] = v;
}

// ---------------------------------------------------------------------------
// k_conv1x1: out = BN(ReLU(W(128x128) @ in + bias)), in/out (B,128,T) f32.
// One wave per 16-column tile, full M=128 computed in two 64-row halves with
// weights staged in (bank-padded) LDS. V_WMMA_F32_16X16X4_F32 path.
// grid = 300 blocks x 128 threads (4 waves); 1200 column tiles total.
// ---------------------------------------------------------------------------
__global__ __launch_bounds__(128) void k_conv1x1(const float* __restrict__ in,
                                                 const float* __restrict__ Wt,
                                                 const float* __restrict__ bias,
                                                 const float* __restrict__ bn,
                                                 float* __restrict__ out) {
  __shared__ float Wlds[64 * 132];
  __shared__ float sc[128], sh[128], bs[128];

  int tid = threadIdx.x;
  if (tid < 128) {
    float g = bn[tid], bb = bn[128 + tid], m = bn[256 + tid], vv = bn[384 + tid];
    float s = g * rsqrtf(vv + EPSF);
    sc[tid] = s; sh[tid] = bb - m * s; bs[tid] = bias[tid];
  }

  int wave = tid >> 5;
  int lane = tid & 31;
  int tileN = blockIdx.x * 4 + wave;          // 0..1199
  int col = tileN * 16 + (lane & 15);         // 0..19199
  int bidx = col / T600;
  int t = col - bidx * T600;
  int koff = (lane >= 16) ? 2 : 0;
  int rowadd = (lane >= 16) ? 8 : 0;
  const float* colbase = in + (size_t)bidx * 128 * T600 + t;
  float* outb = out + (size_t)bidx * 128 * T600 + t;

  for (int half = 0; half < 2; ++half) {
    __syncthreads();
    // stage 64 rows of W into LDS (8192 elems / 128 threads)
    for (int idx = tid; idx < 64 * 128; idx += 128) {
      int o = idx >> 7, c = idx & 127;
      Wlds[o * 132 + c] = Wt[(size_t)(half * 64 + o) * 128 + c];
    }
    __syncthreads();

    v8f acc[4] = {};
#pragma unroll 8
    for (int ks = 0; ks < 32; ++ks) {
      int K = ks * 4 + koff;
      v2f bv;
      bv[0] = colbase[(size_t)K * T600];
      bv[1] = colbase[(size_t)K * T600 + T600];
#pragma unroll
      for (int mt = 0; mt < 4; ++mt) {
        int o = mt * 16 + (lane & 15);
        v2f av;
        av[0] = Wlds[o * 132 + K];
        av[1] = Wlds[o * 132 + K + 1];
        acc[mt] = __builtin_amdgcn_wmma_f32_16x16x4_f32(
            false, av, false, bv, (short)0, acc[mt], false, false);
      }
    }
    // fused bias + ReLU + BN epilogue
#pragma unroll
    for (int mt = 0; mt < 4; ++mt) {
#pragma unroll
      for (int r = 0; r < 8; ++r) {
        int o = half * 64 + mt * 16 + r + rowadd;
        float v = acc[mt][r] + bs[o];
        v = fmaxf(v, 0.f);
        v = v * sc[o] + sh[o];
        outb[(size_t)o * T600] = v;
      }
    }
  }
}

// ---------------------------------------------------------------------------
// k_branch<J>: inner dilated conv branch j of a Bottle2neck.
// GEMM: M=16 (out ch), K=48 (16 ch x 3 taps, K = 16*kh + i), N = 16 t-cols.
// One wave per (b, t-tile); loops over the S=J+2 scale slices accumulating the
// learnable weighted sum directly into out2 channels [16J,16J+16).
// grid = 32*38 blocks x 32 threads.
// ---------------------------------------------------------------------------
template <int J>
__global__ __launch_bounds__(32) void k_branch(const float* __restrict__ bufIn,   // (B,8,16,T)
                                               const float* __restrict__ out1,    // (B,128,T)
                                               float* __restrict__ bufOut,        // (B,8,16,T)
                                               float* __restrict__ out2,          // (B,128,T)
                                               const float* __restrict__ cwj,     // (16,16,3)
                                               const float* __restrict__ cbj,     // (16)
                                               const float* __restrict__ ibnj,    // (4,16)
                                               const float* __restrict__ wsj) {   // (J+2)
  constexpr int S = J + 2;
  int lane = threadIdx.x & 31;
  int blk = blockIdx.x;
  int bidx = blk / 38;
  int t0 = (blk - bidx * 38) * 16;
  int koff = (lane >= 16) ? 2 : 0;
  int rowadd = (lane >= 16) ? 8 : 0;
  int n = lane & 15;

  // preload A operand (weights): 12 k-steps of 4
  v2f aw[12];
  {
    int o = lane & 15;
#pragma unroll
    for (int ks = 0; ks < 12; ++ks) {
      int K0 = ks * 4 + koff;
      int kh0 = K0 >> 4, i0 = K0 & 15;
      int K1 = K0 + 1;
      int kh1 = K1 >> 4, i1 = K1 & 15;
      aw[ks][0] = cwj[(o * 16 + i0) * 3 + kh0];
      aw[ks][1] = cwj[(o * 16 + i1) * 3 + kh1];
    }
  }
  // per-row epilogue params (channel o = r + rowadd)
  float scv[8], shv[8], cbv[8];
#pragma unroll
  for (int r = 0; r < 8; ++r) {
    int o = r + rowadd;
    float g = ibnj[o], bb = ibnj[16 + o], m = ibnj[32 + o], vv = ibnj[48 + o];
    float s = g * rsqrtf(vv + EPSF);
    scv[r] = s; shv[r] = bb - m * s; cbv[r] = cbj[o];
  }

  const float* bb1 = out1 + (size_t)bidx * 128 * T600;
  float wsum[8] = {0.f, 0.f, 0.f, 0.f, 0.f, 0.f, 0.f, 0.f};

#pragma unroll
  for (int s = 0; s < S; ++s) {
    const float* src;
    if (J == 0) {
      src = (s == 0) ? (bb1 + 112 * T600) : bb1;            // [spx7, spx0]
    } else {
      src = (s <= J) ? (bufIn + ((size_t)bidx * 8 + s) * 16 * T600)
                     : (bb1 + (size_t)J * 16 * T600);        // append spx[J]
    }
    v8f acc = {};
#pragma unroll
    for (int ks = 0; ks < 12; ++ks) {
      int K0 = ks * 4 + koff;
      v2f bv;
      {
        int K = K0, kh = K >> 4, ii = K & 15;
        int tt = t0 + n + 2 * kh - 2;
        bv[0] = (tt >= 0 && tt < T600) ? src[ii * T600 + tt] : 0.f;
      }
      {
        int K = K0 + 1, kh = K >> 4, ii = K & 15;
        int tt = t0 + n + 2 * kh - 2;
        bv[1] = (tt >= 0 && tt < T600) ? src[ii * T600 + tt] : 0.f;
      }
      acc = __builtin_amdgcn_wmma_f32_16x16x4_f32(
          false, aw[ks], false, bv, (short)0, acc, false, false);
    }
    // bias + ReLU + BN, store slice, accumulate weighted scale-sum
    float w_s = wsj[s];
    float* dst = bufOut + ((size_t)bidx * 8 + s) * 16 * T600;
    int t = t0 + n;
#pragma unroll
    for (int r = 0; r < 8; ++r) {
      int o = r + rowadd;
      float v = acc[r] + cbv[r];
      v = fmaxf(v, 0.f);
      v = v * scv[r] + shv[r];
      if (t < T600) dst[o * T600 + t] = v;
      wsum[r] += w_s * v;
    }
  }

  int t = t0 + n;
  if (t < T600) {
    float* o2 = out2 + (size_t)bidx * 128 * T600 + (size_t)J * 16 * T600;
#pragma unroll
    for (int r = 0; r < 8; ++r) o2[(r + rowadd) * T600 + t] = wsum[r];
    if (J == 6) {  // tail copy: out2[112:128] = out1[112:128] (spx_inner[7])
      const float* srcT = bb1 + 112 * T600;
      float* dstT = out2 + (size_t)bidx * 128 * T600 + 112 * T600;
#pragma unroll
      for (int r = 0; r < 8; ++r) dstT[(r + rowadd) * T600 + t] = srcT[(r + rowadd) * T600 + t];
    }
  }
}

// ---------------------------------------------------------------------------
// k_se: squeeze (time mean) + 2 tiny FCs -> sigmoid gate. One block per batch.
// ---------------------------------------------------------------------------
__global__ __launch_bounds__(128) void k_se(const float* __restrict__ out3,
                                            const float* __restrict__ se1w,
                                            const float* __restrict__ se1b,
                                            const float* __restrict__ se2w,
                                            const float* __restrict__ se2b,
                                            float* __restrict__ gate) {
  __shared__ float mean[128];
  __shared__ float s1[16];
  int b = blockIdx.x;
  int c = threadIdx.x;
  const float* p = out3 + ((size_t)b * 128 + c) * T600;
  float acc = 0.f;
  for (int t = 0; t < T600; ++t) acc += p[t];
  mean[c] = acc * (1.f / T600);
  __syncthreads();
  if (c < 16) {
    float a = se1b[c];
    for (int k = 0; k < 128; ++k) a += se1w[c * 128 + k] * mean[k];
    s1[c] = fmaxf(a, 0.f);
  }
  __syncthreads();
  float a = se2b[c];
#pragma unroll
  for (int k = 0; k < 16; ++k) a += se2w[c * 16 + k] * s1[k];
  gate[b * 128 + c] = 1.f / (1.f + __expf(-a));
}

// ---------------------------------------------------------------------------
// k_epilogue: carry = obn(ReLU(out3*gate + residual)); fused final-concat
// contribution pooled[b,128i+c] = mean_t ReLU(fbn(carry)).
// grid = 32*128 blocks x 256 threads.
// ---------------------------------------------------------------------------
__global__ __launch_bounds__(256) void k_epilogue(const float* __restrict__ out3,
                                                  const float* __restrict__ gate,
                                                  const float* __restrict__ resid,
                                                  const float* __restrict__ obn,
                                                  const float* __restrict__ fbn,
                                                  float* __restrict__ carry,
                                                  float* __restrict__ pooled,
                                                  int iblk) {
  int b = blockIdx.x >> 7;
  int c = blockIdx.x & 127;
  int tid = threadIdx.x;
  float g = obn[c], bb = obn[128 + c], m = obn[256 + c], vv = obn[384 + c];
  float sc = g * rsqrtf(vv + EPSF), sh = bb - m * sc;
  int fc = iblk * 128 + c;
  float fg = fbn[fc], fb = fbn[1024 + fc], fm = fbn[2048 + fc], fv = fbn[3072 + fc];
  float fsc = fg * rsqrtf(fv + EPSF), fsh = fb - fm * fsc;
  float gt = gate[b * 128 + c];
  const float* p3 = out3 + ((size_t)b * 128 + c) * T600;
  const float* pr = resid + ((size_t)b * 128 + c) * T600;
  float* pc = carry + ((size_t)b * 128 + c) * T600;
  float acc = 0.f;
  for (int t = tid; t < T600; t += 256) {
    float v = p3[t] * gt + pr[t];
    v = fmaxf(v, 0.f) * sc + sh;
    pc[t] = v;
    acc += fmaxf(v * fsc + fsh, 0.f);
  }
  __shared__ float red[256];
  red[tid] = acc;
  __syncthreads();
  for (int ofs = 128; ofs > 0; ofs >>= 1) {
    if (tid < ofs) red[tid] += red[tid + ofs];
    __syncthreads();
  }
  if (tid == 0) pooled[b * 1024 + fc] = red[0] * (1.f / T600);
}

// ---------------------------------------------------------------------------
// k_tail: pooled[b,896+c] = mean_t ReLU(fbn(x[b,896+c,:]))  (spx[7] path)
// ---------------------------------------------------------------------------
__global__ __launch_bounds__(256) void k_tail(const float* __restrict__ x,
                                              const float* __restrict__ fbn,
                                              float* __restrict__ pooled) {
  int b = blockIdx.x >> 7;
  int c = blockIdx.x & 127;
  int tid = threadIdx.x;
  int fc = 896 + c;
  float fg = fbn[fc], fb = fbn[1024 + fc], fm = fbn[2048 + fc], fv = fbn[3072 + fc];
  float fsc = fg * rsqrtf(fv + EPSF), fsh = fb - fm * fsc;
  const float* p = x + ((size_t)b * 1024 + fc) * T600;
  float acc = 0.f;
  for (int t = tid; t < T600; t += 256) acc += fmaxf(p[t] * fsc + fsh, 0.f);
  __shared__ float red[256];
  red[tid] = acc;
  __syncthreads();
  for (int ofs = 128; ofs > 0; ofs >>= 1) {
    if (tid < ofs) red[tid] += red[tid + ofs];
    __syncthreads();
  }
  if (tid == 0) pooled[b * 1024 + fc] = red[0] * (1.f / T600);
}

// ---------------------------------------------------------------------------
// k_fc: (B,1024) @ (2,1024)^T + bias -> (B,2)
// ---------------------------------------------------------------------------
__global__ __launch_bounds__(64) void k_fc(const float* __restrict__ pooled,
                                           const float* __restrict__ fcw,
                                           const float* __restrict__ fcb,
                                           float* __restrict__ out) {
  int idx = threadIdx.x;
  if (idx >= 64) return;
  int b = idx >> 1, nn = idx & 1;
  float a = fcb[nn];
  const float* p = pooled + b * 1024;
  for (int c = 0; c < 1024; ++c) a += p[c] * fcw[nn * 1024 + c];
  out[b * 2 + nn] = a;
}

// ---------------------------------------------------------------------------
extern "C" void kernel_launch(void* const* d_in, const int* in_sizes, int n_in,
                              void* d_out, int out_size, void* d_ws, size_t ws_size,
                              hipStream_t stream) {
  (void)in_sizes; (void)n_in; (void)out_size; (void)ws_size;
  const float* x    = (const float*)d_in[0];
  const float* w1   = (const float*)d_in[1];
  const float* b1   = (const float*)d_in[2];
  const float* bn1  = (const float*)d_in[3];
  const float* cw   = (const float*)d_in[4];
  const float* cb   = (const float*)d_in[5];
  const float* ibn  = (const float*)d_in[6];
  const float* wsj_[7] = {(const float*)d_in[7],  (const float*)d_in[8],
                          (const float*)d_in[9],  (const float*)d_in[10],
                          (const float*)d_in[11], (const float*)d_in[12],
                          (const float*)d_in[13]};
  const float* w3   = (const float*)d_in[14];
  const float* b3   = (const float*)d_in[15];
  const float* bn3  = (const float*)d_in[16];
  const float* se1w = (const float*)d_in[17];
  const float* se1b = (const float*)d_in[18];
  const float* se2w = (const float*)d_in[19];
  const float* se2b = (const float*)d_in[20];
  const float* obn  = (const float*)d_in[21];
  const float* fbn  = (const float*)d_in[22];
  const float* fcw  = (const float*)d_in[23];
  const float* fcb  = (const float*)d_in[24];

  const size_t NB = (size_t)NBATCH * 128 * T600;
  float* wsf    = (float*)d_ws;
  float* x_in   = wsf;
  float* out1   = x_in + NB;
  float* spA    = out1 + NB;
  float* spB    = spA + NB;
  float* out2   = spB + NB;
  float* out3   = out2 + NB;
  float* carry  = out3 + NB;
  float* gate   = carry + NB;
  float* pooled = gate + NBATCH * 128;

  for (int i = 0; i < 7; ++i) {
    k_prep<<<dim3(9600), dim3(256), 0, stream>>>(x, carry, x_in, i, (i > 0) ? 1 : 0);
    k_conv1x1<<<dim3(300), dim3(128), 0, stream>>>(
        x_in, w1 + (size_t)i * 16384, b1 + i * 128, bn1 + i * 512, out1);
    for (int j = 0; j < 7; ++j) {
      const float* cwj  = cw + (size_t)(i * 7 + j) * 768;
      const float* cbj  = cb + (size_t)(i * 7 + j) * 16;
      const float* ibnj = ibn + (size_t)(i * 7 + j) * 64;
      const float* wj   = wsj_[j] + (size_t)i * (j + 2);
      float* bo = (j & 1) ? spB : spA;
      const float* bi = (j & 1) ? spA : spB;
      switch (j) {
        case 0: k_branch<0><<<dim3(1216), dim3(32), 0, stream>>>(bi, out1, bo, out2, cwj, cbj, ibnj, wj); break;
        case 1: k_branch<1><<<dim3(1216), dim3(32), 0, stream>>>(bi, out1, bo, out2, cwj, cbj, ibnj, wj); break;
        case 2: k_branch<2><<<dim3(1216), dim3(32), 0, stream>>>(bi, out1, bo, out2, cwj, cbj, ibnj, wj); break;
        case 3: k_branch<3><<<dim3(1216), dim3(32), 0, stream>>>(bi, out1, bo, out2, cwj, cbj, ibnj, wj); break;
        case 4: k_branch<4><<<dim3(1216), dim3(32), 0, stream>>>(bi, out1, bo, out2, cwj, cbj, ibnj, wj); break;
        case 5: k_branch<5><<<dim3(1216), dim3(32), 0, stream>>>(bi, out1, bo, out2, cwj, cbj, ibnj, wj); break;
        case 6: k_branch<6><<<dim3(1216), dim3(32), 0, stream>>>(bi, out1, bo, out2, cwj, cbj, ibnj, wj); break;
      }
    }
    k_conv1x1<<<dim3(300), dim3(128), 0, stream>>>(
        out2, w3 + (size_t)i * 16384, b3 + i * 128, bn3 + i * 512, out3);
    k_se<<<dim3(32), dim3(128), 0, stream>>>(
        out3, se1w + (size_t)i * 2048, se1b + i * 16, se2w + (size_t)i * 2048,
        se2b + i * 128, gate);
    k_epilogue<<<dim3(4096), dim3(256), 0, stream>>>(
        out3, gate, x_in, obn + i * 512, fbn, carry, pooled, i);
  }
  k_tail<<<dim3(4096), dim3(256), 0, stream>>>(x, fbn, pooled);
  k_fc<<<dim3(1), dim3(64), 0, stream>>>(pooled, fcw, fcb, (float*)d_out);
}